// Net_55448027792023
// MI455X (gfx1250) — compile-verified
//
#include <hip/hip_runtime.h>
#include <math.h>

// ---------------- problem constants (match reference) ----------------
#define NNODES 100000
#define NEDGES 3200000
#define ETOT   (NEDGES + NNODES)   // edges + self loops
#define FIN    1433
#define HDIM   16
#define NCLS   7
#define EPS    1e-12f

typedef __attribute__((ext_vector_type(2))) float v2f;
typedef __attribute__((ext_vector_type(8))) float v8f;

// =====================================================================
// Kernel 1: h = relu(x @ W1 + b1)   via V_WMMA_F32_16X16X4_F32
// One wave per 16-row tile. N = 100000 = 6250 tiles exactly.
// Launch: 625 blocks x 320 threads = 6250 waves (no partial waves,
// EXEC is all-ones for every WMMA as required).
// A (16x4 f32) lane layout: lanes 0-15 -> K=(0,1), lanes 16-31 -> K=(2,3),
//   M = lane&15, VGPR0=K_even, VGPR1=K_odd.
// B (4x16 f32) mirrors A with N = lane&15.
// D: vgpr r -> M = r + 8*(lane>>4), N = lane&15.
// =====================================================================
__global__ void agnn_gemm_relu_wmma(const float* __restrict__ x,
                                    const float* __restrict__ W1,
                                    const float* __restrict__ b1,
                                    float* __restrict__ h)
{
    const int wave = (blockIdx.x * blockDim.x + threadIdx.x) >> 5;
    const int lane = threadIdx.x & 31;
    const int m0   = wave * 16;
    const int n    = lane & 15;
    const int kb   = (lane >> 4) << 1;          // 0 or 2
    const float* __restrict__ xrow = x + (size_t)(m0 + n) * FIN; // M = lane&15

    v8f acc = {};
    int k0 = 0;
    #pragma unroll 2
    for (; k0 + 4 <= FIN; k0 += 4) {
        __builtin_prefetch(xrow + k0 + 128, 0, 1);   // global_prefetch_b8
        v2f a, b;
        a.x = xrow[k0 + kb];
        a.y = xrow[k0 + kb + 1];
        b.x = W1[(k0 + kb) * HDIM + n];
        b.y = W1[(k0 + kb + 1) * HDIM + n];
        acc = __builtin_amdgcn_wmma_f32_16x16x4_f32(
                  false, a, false, b, (short)0, acc, false, false);
    }
    // K tail: 1433 % 4 == 1  -> single column k = 1432, zero-padded
    {
        v2f a = {}, b = {};
        if (kb == 0) {
            a.x = xrow[FIN - 1];
            b.x = W1[(FIN - 1) * HDIM + n];
        }
        acc = __builtin_amdgcn_wmma_f32_16x16x4_f32(
                  false, a, false, b, (short)0, acc, false, false);
    }

    const float bias = b1[n];
    const int   g    = (lane >> 4) * 8;
    #pragma unroll
    for (int r = 0; r < 8; ++r) {
        float v = acc[r] + bias;
        h[(size_t)(m0 + g + r) * HDIM + n] = v > 0.0f ? v : 0.0f;
    }
}

// =====================================================================
// Kernel 2: hn = h * rsqrt(sum(h*h) + eps)   (one thread per node)
// =====================================================================
__global__ void agnn_normalize(const float* __restrict__ h,
                               float* __restrict__ hn)
{
    int i = blockIdx.x * blockDim.x + threadIdx.x;
    if (i >= NNODES) return;
    const float4* __restrict__ p = (const float4*)(h + (size_t)i * HDIM);
    float4 a = p[0], b = p[1], c = p[2], d = p[3];
    float ss = a.x*a.x + a.y*a.y + a.z*a.z + a.w*a.w
             + b.x*b.x + b.y*b.y + b.z*b.z + b.w*b.w
             + c.x*c.x + c.y*c.y + c.z*c.z + c.w*c.w
             + d.x*d.x + d.y*d.y + d.z*d.z + d.w*d.w;
    float rs = rsqrtf(ss + EPS);
    float4* __restrict__ q = (float4*)(hn + (size_t)i * HDIM);
    a.x*=rs; a.y*=rs; a.z*=rs; a.w*=rs;
    b.x*=rs; b.y*=rs; b.z*=rs; b.w*=rs;
    c.x*=rs; c.y*=rs; c.z*=rs; c.w*=rs;
    d.x*=rs; d.y*=rs; d.z*=rs; d.w*=rs;
    q[0]=a; q[1]=b; q[2]=c; q[3]=d;
}

// =====================================================================
// Kernel 3: init per-prop state: m = -inf, s = 0, out = 0
// =====================================================================
__global__ void agnn_init(float* __restrict__ m, float* __restrict__ s,
                          float* __restrict__ out)
{
    int i = blockIdx.x * blockDim.x + threadIdx.x;
    if (i < NNODES) { m[i] = -INFINITY; s[i] = 0.0f; }
    if (i < NNODES * HDIM) out[i] = 0.0f;
}

// ---- float atomic max via int punning (valid for all finite floats) ----
__device__ __forceinline__ void atomic_max_float(float* addr, float val)
{
    if (val >= 0.0f) atomicMax((int*)addr, __float_as_int(val));
    else             atomicMin((unsigned int*)addr, __float_as_uint(val));
}

__device__ __forceinline__ void edge_endpoints(const int* __restrict__ ei,
                                               int e, int& src, int& dst)
{
    if (e < NEDGES) { src = ei[e]; dst = ei[NEDGES + e]; }
    else            { src = dst = e - NEDGES; }          // self loop
}

// =====================================================================
// Pass A: logits[e] = beta * cos(hn[src], hn[dst]); segment-max into m
// beta_ptr == nullptr  ->  beta = 1.0 (prop 1)
// =====================================================================
__global__ void agnn_edge_logits(const int* __restrict__ ei,
                                 const float* __restrict__ hn,
                                 const float* __restrict__ beta_ptr,
                                 float* __restrict__ lg,
                                 float* __restrict__ m)
{
    int e = blockIdx.x * blockDim.x + threadIdx.x;
    if (e >= ETOT) return;
    int src, dst;
    edge_endpoints(ei, e, src, dst);
    const float4* __restrict__ ps = (const float4*)(hn + (size_t)src * HDIM);
    const float4* __restrict__ pd = (const float4*)(hn + (size_t)dst * HDIM);
    float4 a0 = ps[0], a1 = ps[1], a2 = ps[2], a3 = ps[3];
    float4 b0 = pd[0], b1 = pd[1], b2 = pd[2], b3 = pd[3];
    float cosv = a0.x*b0.x + a0.y*b0.y + a0.z*b0.z + a0.w*b0.w
               + a1.x*b1.x + a1.y*b1.y + a1.z*b1.z + a1.w*b1.w
               + a2.x*b2.x + a2.y*b2.y + a2.z*b2.z + a2.w*b2.w
               + a3.x*b3.x + a3.y*b3.y + a3.z*b3.z + a3.w*b3.w;
    float beta  = beta_ptr ? beta_ptr[0] : 1.0f;
    float logit = beta * cosv;
    lg[e] = logit;
    atomic_max_float(m + dst, logit);
}

// =====================================================================
// Pass B: e_val = exp(logit - m[dst]); segment-sum into s  (in-place lg)
// =====================================================================
__global__ void agnn_edge_expsum(const int* __restrict__ ei,
                                 float* __restrict__ lg,
                                 const float* __restrict__ m,
                                 float* __restrict__ s)
{
    int e = blockIdx.x * blockDim.x + threadIdx.x;
    if (e >= ETOT) return;
    int src, dst;
    edge_endpoints(ei, e, src, dst);
    (void)src;
    float v = expf(lg[e] - m[dst]);
    lg[e] = v;
    atomicAdd(s + dst, v);
}

// =====================================================================
// Pass C: out[dst] += (e_val / max(s[dst],eps)) * h[src]
// 16 threads per edge, one feature each.
// =====================================================================
__global__ void agnn_edge_scatter(const int* __restrict__ ei,
                                  const float* __restrict__ lg,
                                  const float* __restrict__ s,
                                  const float* __restrict__ hsrc,
                                  float* __restrict__ out)
{
    int t = blockIdx.x * blockDim.x + threadIdx.x;
    if (t >= ETOT * HDIM) return;
    int e = t >> 4;
    int f = t & 15;
    int src, dst;
    edge_endpoints(ei, e, src, dst);
    float attn = lg[e] / fmaxf(s[dst], EPS);
    atomicAdd(out + (size_t)dst * HDIM + f, attn * hsrc[(size_t)src * HDIM + f]);
}

// =====================================================================
// Head: logits = h @ W2 + b2; log_softmax rows  (one thread per node)
// =====================================================================
__global__ void agnn_head(const float* __restrict__ h,
                          const float* __restrict__ W2,
                          const float* __restrict__ b2,
                          float* __restrict__ out)
{
    int i = blockIdx.x * blockDim.x + threadIdx.x;
    if (i >= NNODES) return;
    float hr[HDIM];
    const float4* __restrict__ p = (const float4*)(h + (size_t)i * HDIM);
    float4 a = p[0], b = p[1], c = p[2], d = p[3];
    hr[0]=a.x; hr[1]=a.y; hr[2]=a.z;  hr[3]=a.w;
    hr[4]=b.x; hr[5]=b.y; hr[6]=b.z;  hr[7]=b.w;
    hr[8]=c.x; hr[9]=c.y; hr[10]=c.z; hr[11]=c.w;
    hr[12]=d.x; hr[13]=d.y; hr[14]=d.z; hr[15]=d.w;

    float z[NCLS];
    #pragma unroll
    for (int cc = 0; cc < NCLS; ++cc) {
        float acc = b2[cc];
        #pragma unroll
        for (int f = 0; f < HDIM; ++f) acc += hr[f] * W2[f * NCLS + cc];
        z[cc] = acc;
    }
    float mx = z[0];
    #pragma unroll
    for (int cc = 1; cc < NCLS; ++cc) mx = fmaxf(mx, z[cc]);
    float sum = 0.0f;
    #pragma unroll
    for (int cc = 0; cc < NCLS; ++cc) sum += expf(z[cc] - mx);
    float lse = mx + logf(sum);
    #pragma unroll
    for (int cc = 0; cc < NCLS; ++cc) out[(size_t)i * NCLS + cc] = z[cc] - lse;
}

// =====================================================================
// launcher
// =====================================================================
extern "C" void kernel_launch(void* const* d_in, const int* in_sizes, int n_in,
                              void* d_out, int out_size, void* d_ws, size_t ws_size,
                              hipStream_t stream)
{
    const float* x    = (const float*)d_in[0];
    const int*   ei   = (const int*)  d_in[1];
    const float* W1   = (const float*)d_in[2];
    const float* b1   = (const float*)d_in[3];
    const float* beta = (const float*)d_in[4];
    const float* W2   = (const float*)d_in[5];
    const float* b2   = (const float*)d_in[6];
    float*       out  = (float*)d_out;

    float* ws  = (float*)d_ws;
    float* h   = ws;                            // N*H  (relu out; later prop2 out)
    float* hn  = h  + (size_t)NNODES * HDIM;    // N*H  (normalized feats, reused)
    float* h2  = hn + (size_t)NNODES * HDIM;    // N*H  (prop1 out)
    float* mb  = h2 + (size_t)NNODES * HDIM;    // N    (segment max)
    float* sb  = mb + NNODES;                   // N    (segment sum)
    float* lg  = sb + NNODES;                   // E+N  (logits -> exp vals)

    const int TPB      = 256;
    const int gNode    = (NNODES + TPB - 1) / TPB;              // 391
    const int gInit    = (NNODES * HDIM + TPB - 1) / TPB;       // 6250
    const int gEdge    = (ETOT + TPB - 1) / TPB;                // 12891
    const int gScatter = (ETOT * HDIM + TPB - 1) / TPB;         // 206250

    // (1) h = relu(x @ W1 + b1)   -- WMMA, 6250 waves exactly
    agnn_gemm_relu_wmma<<<625, 320, 0, stream>>>(x, W1, b1, h);

    // (2) prop 1 (beta = 1.0)
    agnn_normalize   <<<gNode,    TPB, 0, stream>>>(h, hn);
    agnn_init        <<<gInit,    TPB, 0, stream>>>(mb, sb, h2);
    agnn_edge_logits <<<gEdge,    TPB, 0, stream>>>(ei, hn, nullptr, lg, mb);
    agnn_edge_expsum <<<gEdge,    TPB, 0, stream>>>(ei, lg, mb, sb);
    agnn_edge_scatter<<<gScatter, TPB, 0, stream>>>(ei, lg, sb, h, h2);

    // (3) prop 2 (learnable beta), result written back into h
    agnn_normalize   <<<gNode,    TPB, 0, stream>>>(h2, hn);
    agnn_init        <<<gInit,    TPB, 0, stream>>>(mb, sb, h);
    agnn_edge_logits <<<gEdge,    TPB, 0, stream>>>(ei, hn, beta, lg, mb);
    agnn_edge_expsum <<<gEdge,    TPB, 0, stream>>>(ei, lg, mb, sb);
    agnn_edge_scatter<<<gScatter, TPB, 0, stream>>>(ei, lg, sb, h2, h);

    // (4) head + log_softmax
    agnn_head<<<gNode, TPB, 0, stream>>>(h, W2, b2, out);
}